// GATLayerVan_5317169512568
// MI455X (gfx1250) — compile-verified
//
#include <hip/hip_runtime.h>
#include <hip/hip_bf16.h>
#include <math.h>

typedef __attribute__((ext_vector_type(2))) float v2f;
typedef __attribute__((ext_vector_type(8))) float v8f;

#define N_NODES 100000
#define N_EDGES 1600000
#define IN_DIM 128
#define OUT_DIM 64
#define NEG_SLOPE 0.01f

// ---------------------------------------------------------------------------
// K1: z = h @ W_fc^T via V_WMMA_F32_16X16X4_F32.
// One wave computes a 16(M) x 64(N) tile, stepping K=128 in chunks of 4.
// A fragment (16x4 f32): lane<16 -> row=lane,   K = k0+0..1 (2 VGPRs)
//                        lane>=16-> row=lane-16,K = k0+2..3
// B fragment (4x16 f32): lane<16 -> col=lane,   K = k0+0..1 ; B[k][n]=W[n][k]
// C/D (16x16 f32): VGPR r -> M = r + (lane>=16 ? 8 : 0), N = lane&15
// ---------------------------------------------------------------------------
__global__ void gat_gemm_wmma(const float* __restrict__ h,
                              const float* __restrict__ W,
                              float* __restrict__ z,
                              int n_nodes) {
    int gtid = blockIdx.x * blockDim.x + threadIdx.x;
    int wave = gtid >> 5;
    int lane = threadIdx.x & 31;
    int m0 = wave * 16;
    if (m0 >= n_nodes) return;            // wave-uniform: EXEC stays all-ones

    int rowA  = m0 + (lane & 15);
    int khalf = (lane >> 4) * 2;          // 0 for lanes 0-15, 2 for lanes 16-31
    int nlane = lane & 15;

    v8f acc0 = {}, acc1 = {}, acc2 = {}, acc3 = {};

    for (int k0 = 0; k0 < IN_DIM; k0 += 4) {
        v2f a = *(const v2f*)(h + (size_t)rowA * IN_DIM + k0 + khalf);

        v2f b0 = *(const v2f*)(W + (size_t)(nlane +  0) * IN_DIM + k0 + khalf);
        v2f b1 = *(const v2f*)(W + (size_t)(nlane + 16) * IN_DIM + k0 + khalf);
        v2f b2 = *(const v2f*)(W + (size_t)(nlane + 32) * IN_DIM + k0 + khalf);
        v2f b3 = *(const v2f*)(W + (size_t)(nlane + 48) * IN_DIM + k0 + khalf);

        acc0 = __builtin_amdgcn_wmma_f32_16x16x4_f32(false, a, false, b0, (short)0, acc0, false, false);
        acc1 = __builtin_amdgcn_wmma_f32_16x16x4_f32(false, a, false, b1, (short)0, acc1, false, false);
        acc2 = __builtin_amdgcn_wmma_f32_16x16x4_f32(false, a, false, b2, (short)0, acc2, false, false);
        acc3 = __builtin_amdgcn_wmma_f32_16x16x4_f32(false, a, false, b3, (short)0, acc3, false, false);
    }

    int rsel = (lane >> 4) * 8;
#pragma unroll
    for (int r = 0; r < 8; ++r) {
        size_t ro = (size_t)(m0 + r + rsel) * OUT_DIM;
        z[ro + nlane +  0] = acc0[r];
        z[ro + nlane + 16] = acc1[r];
        z[ro + nlane + 32] = acc2[r];
        z[ro + nlane + 48] = acc3[r];
    }
}

// ---------------------------------------------------------------------------
// K2: per-node attention scalars  s_l[n] = z[n]·a[0:64], s_r[n] = z[n]·a[64:128]
// ---------------------------------------------------------------------------
__global__ void gat_node_scores(const float* __restrict__ z,
                                const float* __restrict__ attn,
                                float* __restrict__ s_l,
                                float* __restrict__ s_r,
                                int n) {
    int i = blockIdx.x * blockDim.x + threadIdx.x;
    if (i >= n) return;
    const float4* zr = (const float4*)(z + (size_t)i * OUT_DIM);
    const float4* al = (const float4*)(attn);
    const float4* ar = (const float4*)(attn + OUT_DIM);
    float sl = 0.f, sr = 0.f;
#pragma unroll
    for (int q = 0; q < OUT_DIM / 4; ++q) {
        float4 zv = zr[q];
        float4 av = al[q];
        float4 bv = ar[q];
        sl += zv.x * av.x + zv.y * av.y + zv.z * av.z + zv.w * av.w;
        sr += zv.x * bv.x + zv.y * bv.y + zv.z * bv.z + zv.w * bv.w;
    }
    s_l[i] = sl;
    s_r[i] = sr;
}

__global__ void gat_zero_i32(int* __restrict__ p, int n) {
    int i = blockIdx.x * blockDim.x + threadIdx.x;
    if (i < n) p[i] = 0;
}

// ---------------------------------------------------------------------------
// K3: edge scores e[i] = leaky_relu(s_l[src] + s_r[dst]); count in-degree
// ---------------------------------------------------------------------------
__global__ void gat_edge_scores(const int* __restrict__ src,
                                const int* __restrict__ dst,
                                const float* __restrict__ s_l,
                                const float* __restrict__ s_r,
                                float* __restrict__ e,
                                int* __restrict__ deg,
                                int ne) {
    int i = blockIdx.x * blockDim.x + threadIdx.x;
    if (i >= ne) return;
    int d = dst[i];
    float v = s_l[src[i]] + s_r[d];
    e[i] = (v > 0.f) ? v : NEG_SLOPE * v;
    atomicAdd(&deg[d], 1);
}

// ---------------------------------------------------------------------------
// K4: single-block exclusive scan of deg -> offs (and cursor copy)
// ---------------------------------------------------------------------------
__global__ void gat_scan_excl(const int* __restrict__ deg,
                              int* __restrict__ offs,
                              int* __restrict__ cursor,
                              int n) {
    __shared__ int s[1024];
    int t = threadIdx.x;
    int run = 0;
    for (int base = 0; base < n; base += 1024) {
        int idx = base + t;
        int v = (idx < n) ? deg[idx] : 0;
        s[t] = v;
        __syncthreads();
        int val = v;
        for (int off = 1; off < 1024; off <<= 1) {
            int tmp = (t >= off) ? s[t - off] : 0;
            __syncthreads();
            val += tmp;
            s[t] = val;
            __syncthreads();
        }
        int excl = run + val - v;
        if (idx < n) { offs[idx] = excl; cursor[idx] = excl; }
        run += s[1023];
        __syncthreads();
    }
    if (t == 0) offs[n] = run;
}

// ---------------------------------------------------------------------------
// K5: counting-sort fill of incoming-edge CSR
// ---------------------------------------------------------------------------
__global__ void gat_fill_csr(const int* __restrict__ dst,
                             int* __restrict__ cursor,
                             int* __restrict__ csr,
                             int ne) {
    int i = blockIdx.x * blockDim.x + threadIdx.x;
    if (i >= ne) return;
    int p = atomicAdd(&cursor[dst[i]], 1);
    csr[p] = i;
}

// ---------------------------------------------------------------------------
// K6: node-centric mean-filter + softmax + weighted gather. One wave per node.
// Lanes own 2 output columns each; scalar passes are lane-redundant (uniform
// addresses broadcast from L0/L2). No float atomics anywhere.
// ---------------------------------------------------------------------------
__global__ void gat_node_aggregate(const float* __restrict__ z,
                                   const float* __restrict__ e,
                                   const int* __restrict__ src,
                                   const int* __restrict__ csr,
                                   const int* __restrict__ offs,
                                   const int* __restrict__ deg,
                                   float* __restrict__ out,
                                   int n) {
    int gtid = blockIdx.x * blockDim.x + threadIdx.x;
    int node = gtid >> 5;
    if (node >= n) return;               // wave-uniform
    int lane = threadIdx.x & 31;
    int c0 = lane * 2;

    int d = deg[node];
    int o = offs[node];

    float accx = 0.f, accy = 0.f;
    if (d > 0) {
        // pass 1: per-node mean of e
        float sum = 0.f;
        for (int j = 0; j < d; ++j) sum += e[csr[o + j]];
        float mean = sum / (float)d;

        // pass 2: max of filtered scores e2 = (e < mean ? 0 : e)
        float m = -1e30f;
        for (int j = 0; j < d; ++j) {
            float ev = e[csr[o + j]];
            float e2 = (ev < mean) ? 0.f : ev;
            m = fmaxf(m, e2);
        }

        // pass 3: fused exp-sum + weighted gather of z[src]
        float denom = 0.f;
        for (int j = 0; j < d; ++j) {
            int ei = csr[o + j];
            float ev = e[ei];
            float e2 = (ev < mean) ? 0.f : ev;
            float ex = __expf(e2 - m);
            denom += ex;
            const float2 zv = *(const float2*)(z + (size_t)src[ei] * OUT_DIM + c0);
            accx += ex * zv.x;
            accy += ex * zv.y;
        }
        float inv = 1.f / denom;
        accx *= inv;
        accy *= inv;
    }
    float2 r;
    r.x = accx;
    r.y = accy;
    *(float2*)(out + (size_t)node * OUT_DIM + c0) = r;
}

// ---------------------------------------------------------------------------
extern "C" void kernel_launch(void* const* d_in, const int* in_sizes, int n_in,
                              void* d_out, int out_size, void* d_ws, size_t ws_size,
                              hipStream_t stream) {
    const float* h      = (const float*)d_in[0];
    const int*   src    = (const int*)d_in[1];
    const int*   dst    = (const int*)d_in[2];
    const float* W_fc   = (const float*)d_in[3];
    const float* W_attn = (const float*)d_in[4];
    float* out = (float*)d_out;

    // workspace carve-up (16B aligned regions)
    char* ws = (char*)d_ws;
    size_t off = 0;
    auto carve = [&](size_t bytes) {
        void* p = ws + off;
        off += (bytes + 15) & ~(size_t)15;
        return p;
    };
    float* z      = (float*)carve((size_t)N_NODES * OUT_DIM * sizeof(float));
    float* s_l    = (float*)carve((size_t)N_NODES * sizeof(float));
    float* s_r    = (float*)carve((size_t)N_NODES * sizeof(float));
    float* e      = (float*)carve((size_t)N_EDGES * sizeof(float));
    int*   deg    = (int*)carve((size_t)N_NODES * sizeof(int));
    int*   offs   = (int*)carve((size_t)(N_NODES + 1) * sizeof(int));
    int*   cursor = (int*)carve((size_t)N_NODES * sizeof(int));
    int*   csr    = (int*)carve((size_t)N_EDGES * sizeof(int));
    (void)ws_size;

    const int B = 256;

    // K1: GEMM (6250 waves of 16 rows; 8 waves/block)
    int gemm_waves  = N_NODES / 16;                 // 100000/16 = 6250
    int gemm_blocks = (gemm_waves + 7) / 8;
    gat_gemm_wmma<<<gemm_blocks, B, 0, stream>>>(h, W_fc, z, N_NODES);

    // K2: per-node attention scalars
    gat_node_scores<<<(N_NODES + B - 1) / B, B, 0, stream>>>(z, W_attn, s_l, s_r, N_NODES);

    // K3: zero degree, then edge scores + degree count
    gat_zero_i32<<<(N_NODES + B - 1) / B, B, 0, stream>>>(deg, N_NODES);
    gat_edge_scores<<<(N_EDGES + B - 1) / B, B, 0, stream>>>(src, dst, s_l, s_r, e, deg, N_EDGES);

    // K4: exclusive scan of degrees
    gat_scan_excl<<<1, 1024, 0, stream>>>(deg, offs, cursor, N_NODES);

    // K5: CSR fill
    gat_fill_csr<<<(N_EDGES + B - 1) / B, B, 0, stream>>>(dst, cursor, csr, N_EDGES);

    // K6: node-centric softmax aggregation (1 wave per node, 8 waves/block)
    int agg_blocks = (N_NODES + 7) / 8;
    gat_node_aggregate<<<agg_blocks, B, 0, stream>>>(z, e, src, csr, offs, deg, out, N_NODES);
}